// GCN_Decoder_4853313044733
// MI455X (gfx1250) — compile-verified
//
#include <hip/hip_runtime.h>
#include <hip/hip_bf16.h>
#include <stdint.h>

typedef float v2f __attribute__((ext_vector_type(2)));
typedef float v8f __attribute__((ext_vector_type(8)));

#define HDIM 128

// ---------------- utility kernels ----------------

__global__ __launch_bounds__(256)
void zero_f32(float* __restrict__ p, int n) {
  int i = blockIdx.x * blockDim.x + threadIdx.x;
  int stride = gridDim.x * blockDim.x;
  for (; i < n; i += stride) p[i] = 0.0f;
}

__global__ __launch_bounds__(256)
void count_deg(const int* __restrict__ idx, float* __restrict__ deg, int n) {
  int i = blockIdx.x * blockDim.x + threadIdx.x;
  if (i < n) atomicAdd(&deg[idx[i]], 1.0f);
}

__global__ __launch_bounds__(256)
void finalize_norm(float* __restrict__ deg, int n) {
  int i = blockIdx.x * blockDim.x + threadIdx.x;
  if (i < n) deg[i] = rsqrtf(fmaxf(deg[i], 1.0f));
}

__global__ __launch_bounds__(128)
void bias_sum_k(const float* __restrict__ conv_b, float* __restrict__ bias_sum,
                int LH, int Rr, int H) {
  int i = blockIdx.x * blockDim.x + threadIdx.x;
  if (i < LH) {
    int l = i / H, h = i % H;
    float s = 0.0f;
    for (int r = 0; r < Rr; ++r) s += conv_b[(size_t)(l * Rr + r) * H + h];
    bias_sum[i] = s;
  }
}

// ---------------- edge scatter: h[dst] += m[src] * norm_in[dst] ----------------
// 32 lanes per edge, each lane handles 4 consecutive floats (float4 of 128).

__global__ __launch_bounds__(256)
void scatter_edges(const float* __restrict__ m, const int* __restrict__ src,
                   const int* __restrict__ dst, const float* __restrict__ norm_in,
                   float* __restrict__ h, int E) {
  int g = blockIdx.x * blockDim.x + threadIdx.x;
  int edge = g >> 5;
  if (edge >= E) return;
  int sub = (g & 31) << 2;
  int s = src[edge];
  int d = dst[edge];
  float w = norm_in[d];
  const float4 v = *reinterpret_cast<const float4*>(m + (size_t)s * HDIM + sub);
  float* hp = h + (size_t)d * HDIM + sub;
  atomicAdd(hp + 0, v.x * w);
  atomicAdd(hp + 1, v.y * w);
  atomicAdd(hp + 2, v.z * w);
  atomicAdd(hp + 3, v.w * w);
}

// ---------------- WMMA GEMM (fp32, V_WMMA_F32_16X16X4_F32) ----------------
// C[M x Nc] = epilogue( (A[M x 128] (*row_scale) (+a_bias)) @ B[128 x Nc] + c_bias )
// Block: 256 threads = 8 waves; each wave owns a 16-row strip; B chunk (128x128)
// staged in LDS; A fragments cached in registers (64 VGPRs/lane).

__global__ __launch_bounds__(256)
void wmma_gemm(const float* __restrict__ A, const float* __restrict__ B,
               float* __restrict__ C, int M, int Nc,
               const float* __restrict__ row_scale,   // [M]  or null
               const float* __restrict__ a_bias,      // [128] or null (added to A)
               const float* __restrict__ c_bias,      // [Nc] or null
               const float* __restrict__ bn_g,        // non-null => relu + BN epilogue
               const float* __restrict__ bn_b,
               const float* __restrict__ bn_m,
               const float* __restrict__ bn_v) {
  __shared__ float Bs[HDIM][HDIM];   // 64 KB

  const int tid  = threadIdx.x;
  const int wave = tid >> 5;
  const int lane = tid & 31;
  const int half = lane >> 4;        // 0: lanes 0-15, 1: lanes 16-31
  const int mrow = lane & 15;
  const int row0 = blockIdx.x * 128 + wave * 16;
  const int row  = row0 + mrow;

  const bool row_ok = (row < M);
  const int  rowc   = row_ok ? row : 0;
  float rs;
  if (row_scale) rs = row_ok ? row_scale[row] : 0.0f;
  else           rs = row_ok ? 1.0f : 0.0f;

  // A fragment cache: frag f covers k = 4f + 2*half + {0,1}  (ISA 16x4 f32 layout)
  const float* Arow = A + (size_t)rowc * HDIM;
  v2f afrag[32];
  if (a_bias) {
#pragma unroll
    for (int f = 0; f < 32; ++f) {
      int k = 4 * f + 2 * half;
      float2 av = *reinterpret_cast<const float2*>(Arow + k);
      afrag[f].x = (av.x + a_bias[k])     * rs;
      afrag[f].y = (av.y + a_bias[k + 1]) * rs;
    }
  } else {
#pragma unroll
    for (int f = 0; f < 32; ++f) {
      int k = 4 * f + 2 * half;
      float2 av = *reinterpret_cast<const float2*>(Arow + k);
      afrag[f].x = av.x * rs;
      afrag[f].y = av.y * rs;
    }
  }

  for (int c0 = 0; c0 < Nc; c0 += HDIM) {
    __syncthreads();   // protect previous chunk's Bs reads
    // cooperative load of B chunk: 128x128 floats, float4 per thread x16
#pragma unroll
    for (int it = 0; it < 16; ++it) {
      int i  = it * 256 + tid;        // float4 index 0..4095
      int r  = i >> 5;                // B row 0..127
      int c4 = (i & 31) << 2;         // B col 0..124
      float4 v = *reinterpret_cast<const float4*>(B + (size_t)r * Nc + c0 + c4);
      Bs[r][c4 + 0] = v.x; Bs[r][c4 + 1] = v.y;
      Bs[r][c4 + 2] = v.z; Bs[r][c4 + 3] = v.w;
    }
    __syncthreads();

    for (int nt = 0; nt < 8; ++nt) {
      const int n0   = nt * 16;
      const int colB = n0 + mrow;
      v8f acc = {0.0f, 0.0f, 0.0f, 0.0f, 0.0f, 0.0f, 0.0f, 0.0f};
#pragma unroll
      for (int f = 0; f < 32; ++f) {
        int k = 4 * f + 2 * half;     // B 4x16: VGPR0 rows {k, k+2}, VGPR1 rows {k+1, k+3}
        v2f b;
        b.x = Bs[k][colB];
        b.y = Bs[k + 1][colB];
        acc = __builtin_amdgcn_wmma_f32_16x16x4_f32(
            false, afrag[f], false, b, (short)0, acc, false, false);
      }

      // epilogue: lane owns column col for rows {j + 8*half}
      const int col = c0 + n0 + mrow;
      const float cb = c_bias ? c_bias[col] : 0.0f;
      const bool bn = (bn_g != nullptr);
      float s = 1.0f, t = 0.0f;
      if (bn) {
        float inv = rsqrtf(bn_v[col] + 1e-5f);
        s = bn_g[col] * inv;
        t = bn_b[col] - bn_m[col] * s;
      }
#pragma unroll
      for (int j = 0; j < 8; ++j) {
        int r = row0 + j + 8 * half;
        if (r < M) {
          float z = acc[j] + cb;
          if (bn) z = fmaxf(z, 0.0f) * s + t;
          C[(size_t)r * Nc + col] = z;
        }
      }
    }
  }
}

// ---------------- orchestration ----------------

extern "C" void kernel_launch(void* const* d_in, const int* in_sizes, int n_in,
                              void* d_out, int out_size, void* d_ws, size_t ws_size,
                              hipStream_t stream) {
  const float* x      = (const float*)d_in[0];
  const int*   e_src  = (const int*)d_in[1];
  const int*   e_dst  = (const int*)d_in[2];
  const float* conv_W = (const float*)d_in[3];
  const float* conv_b = (const float*)d_in[4];
  const float* fc_W   = (const float*)d_in[5];
  const float* fc_b   = (const float*)d_in[6];
  const float* fc_Wl  = (const float*)d_in[7];
  const float* fc_bl  = (const float*)d_in[8];
  const float* bn_g   = (const float*)d_in[9];
  const float* bn_b   = (const float*)d_in[10];
  const float* bn_m   = (const float*)d_in[11];
  const float* bn_v   = (const float*)d_in[12];
  float* out = (float*)d_out;

  const int H  = HDIM;
  const int Lr = 3;
  const int Rr = 3;
  const int N  = in_sizes[0] / H;   // 50000
  const int E  = in_sizes[1] / Rr;  // 800000
  const int D  = in_sizes[8];       // 512

  // carve workspace (256B aligned chunks)
  uintptr_t base = (uintptr_t)d_ws;
  auto carve = [&](size_t bytes) -> float* {
    uintptr_t p = (base + 255) & ~(uintptr_t)255;
    base = p + bytes;
    return (float*)p;
  };
  float* norm_out = carve((size_t)Rr * N * 4);
  float* norm_in  = carve((size_t)Rr * N * 4);
  float* bias_sum = carve((size_t)Lr * H * 4);
  float* mbuf     = carve((size_t)N * H * 4);
  float* hbuf     = carve((size_t)N * H * 4);
  float* xbuf     = carve((size_t)N * H * 4);
  (void)ws_size; (void)n_in; (void)out_size;

  dim3 blk(256);
  const int zgrid = 2048;
  zero_f32<<<zgrid, blk, 0, stream>>>(norm_out, Rr * N);
  zero_f32<<<zgrid, blk, 0, stream>>>(norm_in,  Rr * N);

  const int egrid = (E + 255) / 256;
  for (int r = 0; r < Rr; ++r) {
    count_deg<<<egrid, blk, 0, stream>>>(e_src + (size_t)r * E, norm_out + (size_t)r * N, E);
    count_deg<<<egrid, blk, 0, stream>>>(e_dst + (size_t)r * E, norm_in  + (size_t)r * N, E);
  }
  const int ngrid = (Rr * N + 255) / 256;
  finalize_norm<<<ngrid, blk, 0, stream>>>(norm_out, Rr * N);
  finalize_norm<<<ngrid, blk, 0, stream>>>(norm_in,  Rr * N);
  bias_sum_k<<<(Lr * H + 127) / 128, 128, 0, stream>>>(conv_b, bias_sum, Lr * H, Rr, H);

  const int ggrid = (N + 127) / 128;                        // 391 blocks
  const int sgrid = (int)(((size_t)E * 32 + 255) / 256);    // 100000 blocks

  const float* xin = x;
  for (int l = 0; l < Lr; ++l) {
    zero_f32<<<zgrid, blk, 0, stream>>>(hbuf, N * H);
    for (int r = 0; r < Rr; ++r) {
      // m = (x * norm_out_r) @ conv_W[l][r]
      wmma_gemm<<<ggrid, blk, 0, stream>>>(
          xin, conv_W + (size_t)(l * Rr + r) * H * H, mbuf, N, H,
          norm_out + (size_t)r * N, nullptr, nullptr,
          nullptr, nullptr, nullptr, nullptr);
      // h[dst] += m[src] * norm_in_r[dst]
      scatter_edges<<<sgrid, blk, 0, stream>>>(
          mbuf, e_src + (size_t)r * E, e_dst + (size_t)r * E,
          norm_in + (size_t)r * N, hbuf, E);
    }
    if (l < Lr - 1) {
      // x_next = BN(relu((h + bias_sum_l) @ fc_W[l] + fc_b[l]))
      wmma_gemm<<<ggrid, blk, 0, stream>>>(
          hbuf, fc_W + (size_t)l * H * H, xbuf, N, H,
          nullptr, bias_sum + (size_t)l * H, fc_b + (size_t)l * H,
          bn_g + (size_t)l * H, bn_b + (size_t)l * H,
          bn_m + (size_t)l * H, bn_v + (size_t)l * H);
      xin = xbuf;
    } else {
      // out = (h + bias_sum_last) @ fc_W_last + fc_b_last
      wmma_gemm<<<ggrid, blk, 0, stream>>>(
          hbuf, fc_Wl, out, N, D,
          nullptr, bias_sum + (size_t)l * H, fc_bl,
          nullptr, nullptr, nullptr, nullptr);
    }
  }
}